// NIMSCrossEntropyLoss_48902497632522
// MI455X (gfx1250) — compile-verified
//
#include <hip/hip_runtime.h>

// ----------------------------------------------------------------------------
// Cross-entropy (last-S slice) for N=4, S=4, C=4, H=512, W=512, fp32.
// Memory-bound streaming reduction: B128 coalesced loads, fast exp/log,
// block reduction = LDS gather + single wave32 reduction using
// V_WMMA_F32_16X16X4_F32 (ones x partials = column pair-sums) finished with
// ds_swizzle_b32 xor-swaps. Deterministic two-kernel reduction tree.
// ----------------------------------------------------------------------------

typedef __attribute__((ext_vector_type(2))) float v2f;
typedef __attribute__((ext_vector_type(8))) float v8f;

#ifndef CE_N
#define CE_N 4
#define CE_S 4
#define CE_C 4
#define CE_H 512
#define CE_W 512
#endif

static constexpr int kHW         = CE_H * CE_W;                     // 262144 px per (n,c)
static constexpr int kPixPerThr  = 4;                               // one float4 per class
static constexpr int kBlockThr   = 256;                             // 8 waves (wave32)
static constexpr int kBlocksPerN = kHW / (kBlockThr * kPixPerThr);  // 256
static constexpr int kNumBlocks  = CE_N * kBlocksPerN;              // 1024 partials

// One-instruction xor lane swap via ds_swizzle_b32 group-of-32 mode:
// offset = {0, xor_mask[4:0], or_mask=0, and_mask=0x1f}  (SWAPXn patterns).
template <int XorMask>
__device__ __forceinline__ float swz_xor_add(float v) {
  int j = __builtin_amdgcn_ds_swizzle(__float_as_int(v), (XorMask << 10) | 0x1f);
  return v + __int_as_float(j);
}

// Wave32 sum-reduction; caller guarantees all 32 lanes of the wave are active
// (EXEC[31:0] all ones, required by WMMA).
// WMMA trick: D = A(16x4, all ones) * B(4x16) + 0 -> D[m][n] = sum_k B[k][n].
// Each lane's partial occupies one B slot in column (lane & 15), so d[0] holds
// acc[lane&15] + acc[(lane&15)+16] in every lane (32 -> 16 in one matrix op,
// f32 accumulate). Four ds_swizzle xor-swaps finish the reduction.
__device__ __forceinline__ float wave_reduce_sum(float acc) {
#if __has_builtin(__builtin_amdgcn_wmma_f32_16x16x4_f32)
  v2f a; a.x = 1.0f; a.y = 1.0f;   // entire 16x4 A matrix = 1.0
  v2f b; b.x = acc;  b.y = 0.0f;   // lane partial -> one 4x16 B slot
  v8f c = {};
  v8f d = __builtin_amdgcn_wmma_f32_16x16x4_f32(
      /*neg_a=*/false, a, /*neg_b=*/false, b,
      /*c_mod=*/(short)0, c, /*reuse_a=*/false, /*reuse_b=*/false);
  float s = d[0];                  // column sum for column (lane & 15)
  s = swz_xor_add<1>(s);
  s = swz_xor_add<2>(s);
  s = swz_xor_add<4>(s);
  s = swz_xor_add<8>(s);
  return s;                        // total in all lanes
#else
  acc = swz_xor_add<16>(acc);
  acc = swz_xor_add<8>(acc);
  acc = swz_xor_add<4>(acc);
  acc = swz_xor_add<2>(acc);
  acc = swz_xor_add<1>(acc);
  return acc;
#endif
}

__device__ __forceinline__ float pixel_nll(float v0, float v1, float v2, float v3, int t) {
  float m   = fmaxf(fmaxf(v0, v1), fmaxf(v2, v3));
  float sum = __expf(v0 - m) + __expf(v1 - m) + __expf(v2 - m) + __expf(v3 - m);
  float lse = m + __logf(sum);                       // v_exp_f32 / v_log_f32
  // Branchless two-level bit select of the target class value (t in 0..3):
  // 2 bit-tests + 3 v_cndmask, no exec-mask control flow.
  const bool b0 = (t & 1) != 0;
  const bool b1 = (t & 2) != 0;
  const float lo = b0 ? v1 : v0;
  const float hi = b0 ? v3 : v2;
  const float tv = b1 ? hi : lo;
  return lse - tv;
}

__global__ __launch_bounds__(kBlockThr)
void ce_partial_kernel(const float* __restrict__ preds,
                       const int*   __restrict__ targets,
                       float*       __restrict__ partials) {
  const int tid = threadIdx.x;
  const int b   = blockIdx.x;                 // 0 .. 1023
  const int n   = b >> 8;                     // 256 blocks per sample
  const int hw0 = ((((b & (kBlocksPerN - 1)) << 8) | tid) << 2); // 4 consecutive px

  // preds index: (((n*S + (S-1))*C + c) * HW) + hw ; targets: ((n*S + S-1)*HW) + hw
  const size_t pbase = ((size_t)(n * CE_S + (CE_S - 1)) * CE_C) * (size_t)kHW + (size_t)hw0;
  const size_t tbase = ((size_t)(n * CE_S + (CE_S - 1))) * (size_t)kHW + (size_t)hw0;

  // Four B128 class loads (stride HW between classes) + one B128 target load.
  const float4 x0 = *(const float4*)(preds + pbase);
  const float4 x1 = *(const float4*)(preds + pbase + (size_t)kHW);
  const float4 x2 = *(const float4*)(preds + pbase + (size_t)(2 * kHW));
  const float4 x3 = *(const float4*)(preds + pbase + (size_t)(3 * kHW));
  const int4   t  = *(const int4*)(targets + tbase);

  float acc = 0.0f;                                       // fixed order: deterministic
  acc += pixel_nll(x0.x, x1.x, x2.x, x3.x, t.x);
  acc += pixel_nll(x0.y, x1.y, x2.y, x3.y, t.y);
  acc += pixel_nll(x0.z, x1.z, x2.z, x3.z, t.z);
  acc += pixel_nll(x0.w, x1.w, x2.w, x3.w, t.w);

  __shared__ float lds[kBlockThr];
  lds[tid] = acc;
  __syncthreads();

  // Wave 0 only: 32 lanes all active -> EXEC[31:0] all ones for the WMMA.
  if (tid < 32) {
    float s = lds[tid];
#pragma unroll
    for (int i = 1; i < kBlockThr / 32; ++i)   // fixed order: deterministic
      s += lds[tid + (i << 5)];
    const float tot = wave_reduce_sum(s);
    if (tid == 0) partials[b] = tot;
  }
}

__global__ __launch_bounds__(32)
void ce_finalize_kernel(const float* __restrict__ partials,
                        float*       __restrict__ out) {
  const int l = threadIdx.x;                  // single wave, all 32 lanes active
  float acc = 0.0f;
#pragma unroll
  for (int i = 0; i < kNumBlocks / 32; ++i)   // fixed strided order
    acc += partials[l + (i << 5)];
  const float tot = wave_reduce_sum(acc);     // EXEC all ones
  if (l == 0) out[0] = tot * (1.0f / (float)CE_N);
}

extern "C" void kernel_launch(void* const* d_in, const int* in_sizes, int n_in,
                              void* d_out, int out_size, void* d_ws, size_t ws_size,
                              hipStream_t stream) {
  const float* preds    = (const float*)d_in[0];
  const int*   targets  = (const int*)d_in[1];  // JAX default x64-off -> int32
  float*       partials = (float*)d_ws;         // kNumBlocks floats (4 KB)
  float*       out      = (float*)d_out;

  ce_partial_kernel<<<kNumBlocks, kBlockThr, 0, stream>>>(preds, targets, partials);
  ce_finalize_kernel<<<1, 32, 0, stream>>>(partials, out);
}